// GNN13_27410481283382
// MI455X (gfx1250) — compile-verified
//
#include <hip/hip_runtime.h>
#include <hip/hip_bf16.h>

// ---------------------------------------------------------------------------
// EGCN dual-graph network on MI455X (gfx1250).
// Heavy op: m = A @ x  (A: [B,2048,2048] fp32) -> memory bound on streaming A
// (512 MB across the 4 conv applications, ~22us floor at 23.3 TB/s).
// V_WMMA_F32_16X16X4_F32: one wave32 owns a 16-row tile of A, accumulates a
// 16x16 fp32 C tile (features padded 11->16) over K=2048 (512 WMMAs).
// x is pre-padded/transposed to xT[b][16][2048] so both A and B fragments are
// single unconditional global_load_b64 per lane -> branch-free inner loop.
// ---------------------------------------------------------------------------

typedef float v2f __attribute__((ext_vector_type(2)));
typedef float v8f __attribute__((ext_vector_type(8)));

#define GN_N     2048
#define GN_B     8
#define GN_VF    11
#define GN_HEADS 3

// ---------------------------------------------------------------------------
// Pad + transpose: xT[b][n][k] = (n < 11) ? x[b][k][n] : 0   (n in 0..15)
// 1 MB total; writes coalesced along k.
// ---------------------------------------------------------------------------
__global__ __launch_bounds__(256)
void pad_transpose_kernel(const float* __restrict__ x, float* __restrict__ xT)
{
    const int idx = blockIdx.x * 256 + threadIdx.x;      // over B*16*2048
    const int k = idx & (GN_N - 1);
    const int n = (idx >> 11) & 15;
    const int b = idx >> 15;
    xT[idx] = (n < GN_VF) ? x[((size_t)b * GN_N + k) * GN_VF + n] : 0.f;
}

// ---------------------------------------------------------------------------
// Fused EGCN layer: x_out = tanh((x W1) * (A x W2)) Wo
// Block: 256 threads = 8 waves; wave w computes rows [row0+16w, row0+16w+16).
// ---------------------------------------------------------------------------
template <int F>
__global__ __launch_bounds__(256)
void egcn_kernel(const float* __restrict__ A,
                 const float* __restrict__ x_in,   // [B,N,VF] row-major (epilogue)
                 const float* __restrict__ xT,     // [B,16,N] padded transpose
                 const float* __restrict__ W1,
                 const float* __restrict__ W2,
                 const float* __restrict__ Wo,
                 float* __restrict__ x_out)
{
    __shared__ float ms[128 * 16];   // 8 KB: m tiles for this block's 128 rows

    const int b    = blockIdx.y;
    const int row0 = blockIdx.x * 128;

    const float* __restrict__ Ab  = A    + (size_t)b * GN_N * GN_N;
    const float* __restrict__ xb  = x_in + (size_t)b * GN_N * GN_VF;
    const float* __restrict__ xTb = xT   + (size_t)b * 16 * GN_N;

    const int tid  = threadIdx.x;
    const int lane = tid & 31;
    const int wave = tid >> 5;
    const int mloc = lane & 15;    // M row (A frag) / N col (B frag)
    const int kp   = lane >> 4;    // K-pair selector: K{0,1} vs K{2,3}

    const int rowBase = row0 + wave * 16;

    // A fragment stream: row (rowBase+mloc), K = k + 2*kp (+1)
    const float* __restrict__ arow = Ab + (size_t)(rowBase + mloc) * GN_N + 2 * kp;
    // B fragment stream: feature mloc, K = k + 2*kp (+1)  (contiguous float2)
    const float* __restrict__ brow = xTb + (size_t)mloc * GN_N + 2 * kp;

    const v2f* __restrict__ ap = (const v2f*)arow;   // 8B aligned
    const v2f* __restrict__ bp = (const v2f*)brow;   // 8B aligned

    v8f c = {0.f, 0.f, 0.f, 0.f, 0.f, 0.f, 0.f, 0.f};

    for (int kc = 0; kc < GN_N; kc += 256) {
        // CDNA5 speculative prefetch of the A stream (global_prefetch_b8)
        __builtin_prefetch(arow + kc + 512, 0, 0);
        #pragma unroll 8
        for (int k = kc; k < kc + 256; k += 4) {
            v2f a  = ap[k >> 1];
            v2f bb = bp[k >> 1];
            c = __builtin_amdgcn_wmma_f32_16x16x4_f32(
                    /*neg_a=*/false, a, /*neg_b=*/false, bb,
                    /*c_mod=*/(short)0, c, /*reuse_a=*/false, /*reuse_b=*/false);
        }
    }

    // Spill C tile: VGPR j holds M = j + 8*(lane>=16), N = mloc.
    #pragma unroll
    for (int j = 0; j < 8; ++j) {
        const int mr = j + 8 * kp;
        ms[(wave * 16 + mr) * 16 + mloc] = c[j];
    }
    __syncthreads();

    // Epilogue: gated update, 1 thread per row (128 rows / block).
    if (tid < 128) {
        const int row = row0 + tid;
        float xr[GN_VF], mr[GN_VF], acc[GN_VF];
        #pragma unroll
        for (int v = 0; v < GN_VF; ++v) {
            xr[v]  = xb[(size_t)row * GN_VF + v];
            mr[v]  = ms[tid * 16 + v];
            acc[v] = 0.f;
        }
        for (int f = 0; f < F; ++f) {
            float s1 = 0.f, s2 = 0.f;
            #pragma unroll
            for (int v = 0; v < GN_VF; ++v) {
                s1 += xr[v] * W1[v * F + f];
                s2 += mr[v] * W2[v * F + f];
            }
            const float h = tanhf(s1 * s2);
            #pragma unroll
            for (int v = 0; v < GN_VF; ++v)
                acc[v] += h * Wo[f * GN_VF + v];
        }
        float* __restrict__ orow = x_out + ((size_t)b * GN_N + row) * GN_VF;
        #pragma unroll
        for (int v = 0; v < GN_VF; ++v) orow[v] = acc[v];
    }
}

// ---------------------------------------------------------------------------
// Additive-attention pooling: s_n = exp(tanh(x_n . w_h)) (tanh bounded, no
// max-subtraction needed); rep = (sum s_n x_n) / (sum s_n).
// ---------------------------------------------------------------------------
__global__ __launch_bounds__(256)
void att_kernel(const float* __restrict__ x,
                const float* __restrict__ w,
                float* __restrict__ r,
                int colOff)
{
    __shared__ float red[256 * 12];

    const int b = blockIdx.x;
    const int h = blockIdx.y;
    const int tid = threadIdx.x;

    float wv[GN_VF];
    #pragma unroll
    for (int v = 0; v < GN_VF; ++v) wv[v] = w[h * GN_VF + v];

    float esum = 0.f;
    float rep[GN_VF];
    #pragma unroll
    for (int v = 0; v < GN_VF; ++v) rep[v] = 0.f;

    for (int n = tid; n < GN_N; n += 256) {
        const float* __restrict__ xr = x + ((size_t)b * GN_N + n) * GN_VF;
        float d = 0.f;
        #pragma unroll
        for (int v = 0; v < GN_VF; ++v) d += xr[v] * wv[v];
        const float s = __expf(tanhf(d));
        esum += s;
        #pragma unroll
        for (int v = 0; v < GN_VF; ++v) rep[v] += s * xr[v];
    }

    red[tid * 12] = esum;
    #pragma unroll
    for (int v = 0; v < GN_VF; ++v) red[tid * 12 + 1 + v] = rep[v];
    __syncthreads();

    for (int s = 128; s > 0; s >>= 1) {
        if (tid < s) {
            #pragma unroll
            for (int j = 0; j < 12; ++j)
                red[tid * 12 + j] += red[(tid + s) * 12 + j];
        }
        __syncthreads();
    }

    if (tid == 0) {
        const float inv = 1.f / red[0];
        #pragma unroll
        for (int v = 0; v < GN_VF; ++v)
            r[b * (2 * GN_HEADS * GN_VF) + colOff + h * GN_VF + v] = red[1 + v] * inv;
    }
}

// ---------------------------------------------------------------------------
__global__ void head_kernel(const float* __restrict__ r,
                            const float* __restrict__ W,
                            const float* __restrict__ bias,
                            float* __restrict__ out)
{
    const int b = threadIdx.x;
    if (b < GN_B) {
        float a = bias[0];
        for (int j = 0; j < 2 * GN_HEADS * GN_VF; ++j)
            a += r[b * (2 * GN_HEADS * GN_VF) + j] * W[j];
        out[b] = a;
    }
}

// ---------------------------------------------------------------------------
extern "C" void kernel_launch(void* const* d_in, const int* in_sizes, int n_in,
                              void* d_out, int out_size, void* d_ws, size_t ws_size,
                              hipStream_t stream)
{
    const float* x_int  = (const float*)d_in[0];
    const float* x_nh   = (const float*)d_in[1];
    const float* A_int  = (const float*)d_in[2];
    const float* A_nh   = (const float*)d_in[3];
    const float* c1W1   = (const float*)d_in[4];
    const float* c1W2   = (const float*)d_in[5];
    const float* c1Wo   = (const float*)d_in[6];
    const float* c2W1   = (const float*)d_in[7];
    const float* c2W2   = (const float*)d_in[8];
    const float* c2Wo   = (const float*)d_in[9];
    const float* att_iw = (const float*)d_in[10];
    const float* att_nw = (const float*)d_in[11];
    const float* dW     = (const float*)d_in[12];
    const float* db     = (const float*)d_in[13];
    float* out          = (float*)d_out;

    const size_t NBVF = (size_t)GN_B * GN_N * GN_VF;     // 180224 floats
    float* ws  = (float*)d_ws;
    float* x1i = ws;
    float* x1n = ws + NBVF;
    float* x2i = ws + 2 * NBVF;
    float* x2n = ws + 3 * NBVF;
    float* rep = ws + 4 * NBVF;                          // [B, 66]
    float* xT  = ws + 4 * NBVF + 1024;                   // [B,16,N] = 262144 floats

    const dim3 grid(GN_N / 128, GN_B);
    const dim3 blk(256);
    const int  padBlocks = (GN_B * 16 * GN_N) / 256;     // 1024

    // conv1 (F1 = 32), shared weights on both graphs
    pad_transpose_kernel<<<padBlocks, blk, 0, stream>>>(x_int, xT);
    egcn_kernel<32><<<grid, blk, 0, stream>>>(A_int, x_int, xT, c1W1, c1W2, c1Wo, x1i);
    pad_transpose_kernel<<<padBlocks, blk, 0, stream>>>(x_nh, xT);
    egcn_kernel<32><<<grid, blk, 0, stream>>>(A_nh,  x_nh,  xT, c1W1, c1W2, c1Wo, x1n);
    // conv2 (F2 = 64)
    pad_transpose_kernel<<<padBlocks, blk, 0, stream>>>(x1i, xT);
    egcn_kernel<64><<<grid, blk, 0, stream>>>(A_int, x1i, xT, c2W1, c2W2, c2Wo, x2i);
    pad_transpose_kernel<<<padBlocks, blk, 0, stream>>>(x1n, xT);
    egcn_kernel<64><<<grid, blk, 0, stream>>>(A_nh,  x1n, xT, c2W1, c2W2, c2Wo, x2n);

    // dual self-attention pooling -> rep[B, 66]
    att_kernel<<<dim3(GN_B, GN_HEADS), blk, 0, stream>>>(x2i, att_iw, rep, 0);
    att_kernel<<<dim3(GN_B, GN_HEADS), blk, 0, stream>>>(x2n, att_nw, rep, GN_HEADS * GN_VF);

    // linear head
    head_kernel<<<1, 32, 0, stream>>>(rep, dW, db, out);
}